// MultiHeadAttentionLayer_13391708029125
// MI455X (gfx1250) — compile-verified
//
#include <hip/hip_runtime.h>
#include <hip/hip_bf16.h>

typedef float v2f __attribute__((ext_vector_type(2)));
typedef float v8f __attribute__((ext_vector_type(8)));
typedef float f4  __attribute__((ext_vector_type(4)));

// Problem constants: B=4, N=256, H=8, D=8, HD=64, B*N=1024
#define BN   1024
#define NN   256
#define HDIM 64

static __device__ __forceinline__ v8f wmma_f32_k4(v2f a, v2f b, v8f c) {
  // D = A(16x4 f32) x B(4x16 f32) + C(16x16 f32)
  return __builtin_amdgcn_wmma_f32_16x16x4_f32(
      /*neg_a=*/false, a, /*neg_b=*/false, b,
      /*c_mod=*/(short)0, c, /*reuse_a=*/false, /*reuse_b=*/false);
}

// ---------------------------------------------------------------------------
// Kernel 1: Qh = h@Wq^T, Kh = (h@Wk^T)*D^-0.5, Vh = h@Wv^T  via f32 WMMA.
// One wave per (16-row tile, 16-col tile, matrix). 768 waves total.
// ---------------------------------------------------------------------------
__global__ __launch_bounds__(128)
void proj_wmma_kernel(const float* __restrict__ h,
                      const float* __restrict__ Wq,
                      const float* __restrict__ Wk,
                      const float* __restrict__ Wv,
                      float* __restrict__ Qh,
                      float* __restrict__ Kh,
                      float* __restrict__ Vh) {
  const int w    = blockIdx.x * 4 + (threadIdx.x >> 5);   // 0..767
  const int lane = threadIdx.x & 31;
  const int r16  = lane & 15;
  const int hi   = lane >> 4;
  const int mat   = w >> 8;        // 0:Q 1:K 2:V
  const int rem   = w & 255;
  const int mtile = rem >> 2;      // 0..63 (16 rows each over B*N=1024)
  const int ctile = rem & 3;       // 0..3  (16 cols each over HD=64)

  const float* Wm  = (mat == 0) ? Wq : (mat == 1 ? Wk : Wv);
  float*       out = (mat == 0) ? Qh : (mat == 1 ? Kh : Vh);
  const float scale = (mat == 1) ? 0.35355339059327373f : 1.0f;  // D^-0.5

  const int row = mtile * 16 + r16;  // A: M = lane%16
  const int col = ctile * 16 + r16;  // B: N = lane%16 ; B[k][n] = W[n][k]

  v8f acc = {};
  #pragma unroll
  for (int kb = 0; kb < HDIM; kb += 4) {
    const int k = kb + hi * 2;       // lanes 16-31 hold K-pair {2,3} of the tile
    v2f a  = *(const v2f*)(h  + row * HDIM + k);
    v2f bm = *(const v2f*)(Wm + col * HDIM + k);
    acc = wmma_f32_k4(a, bm, acc);
  }
  #pragma unroll
  for (int r = 0; r < 8; ++r) {      // D: M = r + 8*hi, N = r16
    out[(mtile * 16 + r + 8 * hi) * HDIM + ctile * 16 + r16] = acc[r] * scale;
  }
}

// ---------------------------------------------------------------------------
// Kernel 2: scores[b,i,h,j] = sum_k Qh[b,i,h,k]*Kh[b,j,h,k]  (K=8 -> 2 WMMAs)
// One wave per (b,h,itile,jtile). 4*8*16*16 = 8192 waves.
// ---------------------------------------------------------------------------
__global__ __launch_bounds__(256)
void scores_wmma_kernel(const float* __restrict__ Qh,
                        const float* __restrict__ Kh,
                        float* __restrict__ Sc) {
  const int w    = blockIdx.x * 8 + (threadIdx.x >> 5);   // 0..8191
  const int lane = threadIdx.x & 31;
  const int r16  = lane & 15;
  const int hi   = lane >> 4;
  const int b   = w >> 11;          // /(8*256)
  const int rem = w & 2047;
  const int hh  = rem >> 8;         // head
  const int t   = rem & 255;
  const int it  = t >> 4;
  const int jt  = t & 15;

  const float* qb = Qh + (b * NN + it * 16 + r16) * HDIM + hh * 8 + hi * 2;
  const float* kb = Kh + (b * NN + jt * 16 + r16) * HDIM + hh * 8 + hi * 2;

  v8f acc = {};
  acc = wmma_f32_k4(*(const v2f*)(qb),     *(const v2f*)(kb),     acc);
  acc = wmma_f32_k4(*(const v2f*)(qb + 4), *(const v2f*)(kb + 4), acc);

  #pragma unroll
  for (int r = 0; r < 8; ++r) {
    const int i = it * 16 + r + 8 * hi;
    const int j = jt * 16 + r16;
    Sc[((size_t)(b * NN + i) * 8 + hh) * NN + j] = acc[r];
  }
}

// ---------------------------------------------------------------------------
// Kernel 3: bandwidth-bound streaming pass over e_att/e_value.
// One block per (b,i); thread owns 4 channels (b128 loads) and a j-strip.
// Non-temporal loads keep the 134MB stream out of cache; Vh/Sc/mask reused
// from L2.
// ---------------------------------------------------------------------------
__global__ __launch_bounds__(256)
void attn_stream_kernel(const float* __restrict__ e_att,
                        const float* __restrict__ e_value,
                        const float* __restrict__ mask,
                        const float* __restrict__ Vh,
                        const float* __restrict__ Sc,
                        float* __restrict__ out) {
  __shared__ float num_s[16][64];
  __shared__ float den_s[16][64];

  const int tid   = threadIdx.x;
  const int c0    = (tid & 15) * 4;   // channel group (within one head: c0%8 in {0,4})
  const int strip = tid >> 4;         // j-strip 0..15
  const int hch   = c0 >> 3;          // head of this channel group
  const int i     = blockIdx.x & 255;
  const int b     = blockIdx.x >> 8;

  const float  maski = mask[b * NN + i];
  const size_t ebase = (size_t)(b * NN + i) * NN * HDIM;
  const float* scrow = Sc + ((size_t)(b * NN + i) * 8 + hch) * NN;

  f4 num = {0.f, 0.f, 0.f, 0.f};
  f4 den = {0.f, 0.f, 0.f, 0.f};

  for (int j = strip; j < NN; j += 16) {
    const size_t idx = ebase + (size_t)j * HDIM + c0;
    f4 ea = __builtin_nontemporal_load((const f4*)(e_att   + idx));
    f4 ev = __builtin_nontemporal_load((const f4*)(e_value + idx));
    f4 vh = *(const f4*)(Vh + (size_t)(b * NN + j) * HDIM + c0);
    const float dot = scrow[j];
    const float m   = maski * mask[b * NN + j];
    if (j + 16 < NN) {   // global_prefetch_b8 for next strip
      __builtin_prefetch(e_att   + idx + 16 * HDIM, 0, 0);
      __builtin_prefetch(e_value + idx + 16 * HDIM, 0, 0);
    }
    #pragma unroll
    for (int k = 0; k < 4; ++k) {
      float x = dot + ea[k];
      x = fminf(fmaxf(x, -5.0f), 5.0f);
      float s = __expf(x) * m;      // scores * attn_mask
      den[k] += s;                  // softmax denom uses single mask
      s *= m;                       // dropout path multiplies mask again
      num[k] += s * (vh[k] + ev[k]);
    }
  }

  #pragma unroll
  for (int k = 0; k < 4; ++k) {
    num_s[strip][c0 + k] = num[k];
    den_s[strip][c0 + k] = den[k];
  }
  __syncthreads();

  if (tid < 64) {
    float ns = 0.f, ds = 0.f;
    #pragma unroll
    for (int sgp = 0; sgp < 16; ++sgp) {
      ns += num_s[sgp][tid];
      ds += den_s[sgp][tid];
    }
    out[(size_t)(b * NN + i) * HDIM + tid] = ns / fmaxf(ds, 1e-6f);
  }
}

// ---------------------------------------------------------------------------
extern "C" void kernel_launch(void* const* d_in, const int* in_sizes, int n_in,
                              void* d_out, int out_size, void* d_ws, size_t ws_size,
                              hipStream_t stream) {
  const float* h       = (const float*)d_in[0];
  const float* e_att   = (const float*)d_in[1];
  const float* e_value = (const float*)d_in[2];
  const float* mask    = (const float*)d_in[3];
  const float* Wq      = (const float*)d_in[4];
  const float* Wk      = (const float*)d_in[5];
  const float* Wv      = (const float*)d_in[6];

  float* ws = (float*)d_ws;
  float* Qh = ws;                    // 65536 floats
  float* Kh = ws + 65536;            // 65536 floats (pre-scaled by D^-0.5)
  float* Vh = ws + 131072;           // 65536 floats
  float* Sc = ws + 196608;           // 2,097,152 floats: scores (b,i,h,j)

  proj_wmma_kernel  <<<192, 128, 0, stream>>>(h, Wq, Wk, Wv, Qh, Kh, Vh);
  scores_wmma_kernel<<<1024, 256, 0, stream>>>(Qh, Kh, Sc);
  attn_stream_kernel<<<1024, 256, 0, stream>>>(e_att, e_value, mask, Vh, Sc,
                                               (float*)d_out);
}